// Dense_RBS_state_vector_31963146617513
// MI455X (gfx1250) — compile-verified
//
#include <hip/hip_runtime.h>
#include <math.h>

// Problem constants (match reference): 32 qubits -> d = C(32,2) = 496 basis
// states, 62 RBS gates (i,i+1) for i=0..30, twice. BATCH = 4096.
#define NQ     32
#define DDIM   496      // 31 * 16, 124 * 4
#define LDW    512      // padded row stride for composed operator
#define NGATES 62
#define BATCH  4096     // 256 * 16

typedef __attribute__((ext_vector_type(2))) float v2f;
typedef __attribute__((ext_vector_type(8))) float v8f;

// Lexicographic index of pair (a,b), a<b, over 32 qubits.
__device__ __forceinline__ int pair_idx(int a, int b) {
    return a * 31 - (a * (a - 1)) / 2 + (b - a - 1);
}

// ---------------------------------------------------------------------------
// Kernel 1: compose the 62 Givens-rotation gates into a dense operator.
// One 32-thread block per basis column m: simulate the circuit on e_m with the
// 496-float state in LDS (each gate = 30 disjoint rotations, one per lane).
// Result r[k] = T[k][m]; store transposed-for-GEMM: BT[k*LDW + m] = r[k],
// so BT[k][m] = W[m][k] with W = T^T (the matrix the GEMM contracts over).
// ---------------------------------------------------------------------------
__global__ __launch_bounds__(32)
void rbs_compose_kernel(const float* __restrict__ angles, float* __restrict__ BT) {
    const int m    = blockIdx.x;    // 0..495
    const int lane = threadIdx.x;   // 0..31 (wave32)
    __shared__ float v[DDIM];

    for (int k = lane; k < DDIM; k += 32) v[k] = (k == m) ? 1.0f : 0.0f;
    __syncthreads();

    for (int g = 0; g < NGATES; ++g) {
        const int i = g % 31;       // GATES = [(i,i+1) for i in 0..30] * 2
        const int j = i + 1;
        const float th = angles[g];
        const float ct = cosf(th);
        const float st = sinf(th);
        const int c = lane;         // spectator qubit handled by this lane
        if (c != i && c != j) {
            const int ai = (i < c) ? i : c, bi = (i < c) ? c : i;
            const int aj = (j < c) ? j : c, bj = (j < c) ? c : j;
            const int k1 = pair_idx(ai, bi);   // pair containing i
            const int k2 = pair_idx(aj, bj);   // pair containing j
            const float x = v[k1];
            const float y = v[k2];
            v[k1] = ct * x + st * y;           // +sin into the i-pair
            v[k2] = ct * y - st * x;           // -sin into the j-pair
        }
        __syncthreads();            // gates overlap coordinates -> sync
    }

    for (int k = lane; k < DDIM; k += 32)
        BT[(size_t)k * LDW + m] = v[k];
}

// ---------------------------------------------------------------------------
// Kernel 2: out[4096 x 496] = state[4096 x 496] @ W[496 x 496]
// via V_WMMA_F32_16X16X4_F32. Each wave computes a 64x16 output slab:
// 4 M-tiles sharing one B fragment per K-step (1 B load + 4 A loads feed
// 4 WMMAs -> 1.25 loads/WMMA) with 4 independent accumulator chains to keep
// the matrix pipe from stalling on C->D forwarding.
//
// Fragment layouts (wave32):
//   A 16x4 : lane<16 -> (M=lane,  K=0,1) ; lane>=16 -> (M=lane-16, K=2,3)
//   B 4x16 : lane<16 -> (K=0,1, N=lane) ; lane>=16 -> (K=2,3, N=lane-16)
//     -> with BT[k][m] storage both fragments are one contiguous float2 load.
//   C/D    : VGPR v -> row M=v (lanes 0-15) / M=v+8 (lanes 16-31), N=lane%16
// ---------------------------------------------------------------------------
__global__ __launch_bounds__(256)
void rbs_gemm_wmma_kernel(const float* __restrict__ S,
                          const float* __restrict__ BT,
                          float* __restrict__ out) {
    const int wave = (blockIdx.x * blockDim.x + threadIdx.x) >> 5; // 0..1983
    const int lane = threadIdx.x & 31;
    const int mblk = wave & 63;     // 64 slabs of 64 batch rows
    const int ntile = wave >> 6;    // 31 output-coord tiles (1984 = 31*64)
    const int m0   = mblk * 64;
    const int n0   = ntile * 16;
    const int half = lane >> 4;     // 0: K=0,1   1: K=2,3
    const int l16  = lane & 15;

    const float* a0 = S + (size_t)(m0 +  0 + l16) * DDIM + 2 * half;
    const float* a1 = S + (size_t)(m0 + 16 + l16) * DDIM + 2 * half;
    const float* a2 = S + (size_t)(m0 + 32 + l16) * DDIM + 2 * half;
    const float* a3 = S + (size_t)(m0 + 48 + l16) * DDIM + 2 * half;
    const float* br = BT + (size_t)(n0 + l16) * LDW + 2 * half;

    v8f acc0 = {}, acc1 = {}, acc2 = {}, acc3 = {};
#pragma unroll 2
    for (int k4 = 0; k4 < DDIM; k4 += 4) {
        v2f b  = *(const v2f*)(br + k4);
        v2f fa0 = *(const v2f*)(a0 + k4);
        v2f fa1 = *(const v2f*)(a1 + k4);
        v2f fa2 = *(const v2f*)(a2 + k4);
        v2f fa3 = *(const v2f*)(a3 + k4);
        // (neg_a, A, neg_b, B, c_mod, C, reuse_a, reuse_b)
        acc0 = __builtin_amdgcn_wmma_f32_16x16x4_f32(false, fa0, false, b,
                                                     (short)0, acc0, false, false);
        acc1 = __builtin_amdgcn_wmma_f32_16x16x4_f32(false, fa1, false, b,
                                                     (short)0, acc1, false, false);
        acc2 = __builtin_amdgcn_wmma_f32_16x16x4_f32(false, fa2, false, b,
                                                     (short)0, acc2, false, false);
        acc3 = __builtin_amdgcn_wmma_f32_16x16x4_f32(false, fa3, false, b,
                                                     (short)0, acc3, false, false);
    }

    float* o0 = out + (size_t)(m0 +  0 + 8 * half) * DDIM + n0 + l16;
    float* o1 = out + (size_t)(m0 + 16 + 8 * half) * DDIM + n0 + l16;
    float* o2 = out + (size_t)(m0 + 32 + 8 * half) * DDIM + n0 + l16;
    float* o3 = out + (size_t)(m0 + 48 + 8 * half) * DDIM + n0 + l16;
#pragma unroll
    for (int vv = 0; vv < 8; ++vv) {
        o0[(size_t)vv * DDIM] = acc0[vv];
        o1[(size_t)vv * DDIM] = acc1[vv];
        o2[(size_t)vv * DDIM] = acc2[vv];
        o3[(size_t)vv * DDIM] = acc3[vv];
    }
}

// ---------------------------------------------------------------------------
// Harness entry. d_in[0]=input_state f32[4096*496], d_in[1]=angles f32[62],
// d_in[2]=U (unused: we rebuild the gate action analytically).
// d_ws: needs 496*512*4 = 1,015,808 bytes for the composed operator.
// ---------------------------------------------------------------------------
extern "C" void kernel_launch(void* const* d_in, const int* in_sizes, int n_in,
                              void* d_out, int out_size, void* d_ws, size_t ws_size,
                              hipStream_t stream) {
    (void)in_sizes; (void)n_in; (void)out_size; (void)ws_size;
    const float* state  = (const float*)d_in[0];
    const float* angles = (const float*)d_in[1];
    float* BT  = (float*)d_ws;
    float* out = (float*)d_out;

    // Stage 1: compose circuit into dense operator (one wave per basis column).
    rbs_compose_kernel<<<DDIM, 32, 0, stream>>>(angles, BT);

    // Stage 2: single fp32 WMMA GEMM over 64x16 slabs.
    // 1984 waves = 248 blocks * 8 waves.
    rbs_gemm_wmma_kernel<<<248, 256, 0, stream>>>(state, BT, out);
}